// GCNModel_3332894622171
// MI455X (gfx1250) — compile-verified
//
#include <hip/hip_runtime.h>
#include <math.h>

#define N_NODES 100000
#define N_EDGES 3200000
#define IN_F    1433
#define HID     16
#define N_CLS   7

typedef __attribute__((ext_vector_type(2))) float v2f;
typedef __attribute__((ext_vector_type(8))) float v8f;

// ---------------- utility kernels ----------------
__global__ void k_zero(float* __restrict__ p, int n) {
    int i = blockIdx.x * blockDim.x + threadIdx.x;
    if (i < n) p[i] = 0.0f;
}

__global__ void k_init_deg(float* __restrict__ deg, int n) {
    int i = blockIdx.x * blockDim.x + threadIdx.x;
    if (i < n) deg[i] = 1.0f;   // self-loop contributes 1 to every node's degree
}

__global__ void k_deg_accum(const int* __restrict__ col, float* __restrict__ deg, int e) {
    int i = blockIdx.x * blockDim.x + threadIdx.x;
    if (i < e) atomicAdd(&deg[col[i]], 1.0f);
}

__global__ void k_dinv(float* __restrict__ deg, int n) {
    int i = blockIdx.x * blockDim.x + threadIdx.x;
    if (i < n) deg[i] = rsqrtf(deg[i]);
}

// ---------------- GEMM1: h1 = x @ W1 + b1 via V_WMMA_F32_16X16X4_F32 ----------------
// One wave32 computes a 16x16 tile (16 node rows x HID=16 cols), K=1433 stepped by 4.
__global__ __launch_bounds__(256) void k_gemm1_wmma(const float* __restrict__ x,
                                                    const float* __restrict__ W1,
                                                    const float* __restrict__ b1,
                                                    float* __restrict__ h1) {
    const int gtid = blockIdx.x * blockDim.x + threadIdx.x;
    const int tile = gtid >> 5;             // one 16-row tile per wave
    const int lane = threadIdx.x & 31;
    if (tile >= N_NODES / 16) return;       // wave-uniform exit (EXEC stays all-ones inside)

    const int m    = lane & 15;             // A row / B col / D col index
    const int kOff = (lane >> 4) << 1;      // 0 for lanes 0-15, 2 for lanes 16-31

    const float* __restrict__ arow = x  + (size_t)(tile * 16 + m) * IN_F;
    const float* __restrict__ bcol = W1 + m;   // column n == m of W1 (row-major [IN_F, HID])

    v8f acc = {};
    const int KMAIN = (IN_F / 4) * 4;       // 1432
    #pragma unroll 2
    for (int k = 0; k < KMAIN; k += 4) {
        v2f a, b;
        a.x = arow[k + kOff];
        a.y = arow[k + kOff + 1];
        b.x = bcol[(size_t)(k + kOff)     * HID];
        b.y = bcol[(size_t)(k + kOff + 1) * HID];
        acc = __builtin_amdgcn_wmma_f32_16x16x4_f32(false, a, false, b,
                                                    (short)0, acc, false, false);
    }
    // K tail: only K=1432 is valid (lanes with kOff==0, .x element); rest padded with zeros.
    {
        v2f a = {0.0f, 0.0f};
        v2f b = {0.0f, 0.0f};
        if (kOff == 0) {
            a.x = arow[KMAIN];
            b.x = bcol[(size_t)KMAIN * HID];
        }
        acc = __builtin_amdgcn_wmma_f32_16x16x4_f32(false, a, false, b,
                                                    (short)0, acc, false, false);
    }

    const float bias = b1[m];                       // n == m
    const int rowBase = tile * 16 + ((lane >> 4) << 3);   // +8 for upper half lanes
    float* __restrict__ out = h1 + (size_t)rowBase * HID + m;
    #pragma unroll
    for (int v = 0; v < 8; ++v)
        out[(size_t)v * HID] = acc[v] + bias;
}

// ---------------- edge aggregation: agg[col] += dinv[row]*dinv[col] * h[row] ----------------
// LPE lanes cooperate on one edge so feature gathers/atomics are contiguous.
template <int F, int LPE>
__global__ __launch_bounds__(256) void k_aggregate(const int* __restrict__ row,
                                                   const int* __restrict__ col,
                                                   const float* __restrict__ dinv,
                                                   const float* __restrict__ h,
                                                   float* __restrict__ agg,
                                                   int nEdges) {
    int t = blockIdx.x * blockDim.x + threadIdx.x;
    int e = t / LPE;
    int f = t % LPE;
    if (e >= nEdges || f >= F) return;
    int r = row[e];
    int c = col[e];
    float norm = dinv[r] * dinv[c];
    atomicAdd(&agg[(size_t)c * F + f], norm * h[(size_t)r * F + f]);
}

// ---------------- finalize layer 1: h1a = relu(agg1 + dinv^2 * h1)  (in-place into agg1) ----
__global__ void k_finalize1(float* __restrict__ agg,
                            const float* __restrict__ h1,
                            const float* __restrict__ dinv) {
    int i = blockIdx.x * blockDim.x + threadIdx.x;
    if (i >= N_NODES * HID) return;
    int node = i >> 4;   // HID == 16
    float d = dinv[node];
    float v = agg[i] + d * d * h1[i];
    agg[i] = v > 0.0f ? v : 0.0f;
}

// ---------------- GEMM2: h2 = h1a @ W2 + b2  (K=16, N=7, one thread per node) -------------
__global__ __launch_bounds__(256) void k_gemm2(const float* __restrict__ h1a,
                                               const float* __restrict__ W2,
                                               const float* __restrict__ b2,
                                               float* __restrict__ h2) {
    __shared__ float sW[HID * N_CLS];
    __shared__ float sB[N_CLS];
    if (threadIdx.x < HID * N_CLS) sW[threadIdx.x] = W2[threadIdx.x];
    if (threadIdx.x < N_CLS)       sB[threadIdx.x] = b2[threadIdx.x];
    __syncthreads();

    int i = blockIdx.x * blockDim.x + threadIdx.x;
    if (i >= N_NODES) return;

    float hr[HID];
    const float4* p = (const float4*)(h1a + (size_t)i * HID);   // 64B-aligned rows
    #pragma unroll
    for (int q = 0; q < HID / 4; ++q) {
        float4 v = p[q];
        hr[q * 4 + 0] = v.x; hr[q * 4 + 1] = v.y;
        hr[q * 4 + 2] = v.z; hr[q * 4 + 3] = v.w;
    }
    #pragma unroll
    for (int c = 0; c < N_CLS; ++c) {
        float acc = sB[c];
        #pragma unroll
        for (int k = 0; k < HID; ++k)
            acc = fmaf(hr[k], sW[k * N_CLS + c], acc);
        h2[(size_t)i * N_CLS + c] = acc;
    }
}

// ---------------- finalize layer 2 + log_softmax -> d_out ----------------
__global__ __launch_bounds__(256) void k_final(const float* __restrict__ agg2,
                                               const float* __restrict__ h2,
                                               const float* __restrict__ dinv,
                                               float* __restrict__ out) {
    int i = blockIdx.x * blockDim.x + threadIdx.x;
    if (i >= N_NODES) return;
    float d  = dinv[i];
    float dd = d * d;
    float v[N_CLS];
    float mx = -INFINITY;
    #pragma unroll
    for (int c = 0; c < N_CLS; ++c) {
        float t = agg2[(size_t)i * N_CLS + c] + dd * h2[(size_t)i * N_CLS + c];
        t = t > 0.0f ? t : 0.0f;           // relu
        v[c] = t;
        mx = fmaxf(mx, t);
    }
    float s = 0.0f;
    #pragma unroll
    for (int c = 0; c < N_CLS; ++c) s += __expf(v[c] - mx);
    float lse = __logf(s) + mx;
    #pragma unroll
    for (int c = 0; c < N_CLS; ++c)
        out[(size_t)i * N_CLS + c] = v[c] - lse;
}

// ---------------- launch ----------------
extern "C" void kernel_launch(void* const* d_in, const int* in_sizes, int n_in,
                              void* d_out, int out_size, void* d_ws, size_t ws_size,
                              hipStream_t stream) {
    (void)in_sizes; (void)n_in; (void)out_size; (void)ws_size;

    const float* x  = (const float*)d_in[0];
    const float* W1 = (const float*)d_in[1];
    const float* b1 = (const float*)d_in[2];
    const float* W2 = (const float*)d_in[3];
    const float* b2 = (const float*)d_in[4];
    const int*   ei = (const int*)d_in[5];
    const int* row = ei;              // edge_index[0]
    const int* col = ei + N_EDGES;    // edge_index[1]
    float* out = (float*)d_out;

    // workspace layout (all float32, 256B aligned slices)
    char* ws = (char*)d_ws;
    auto carve = [&](size_t elems) {
        float* p = (float*)ws;
        ws += ((elems * sizeof(float)) + 255) & ~(size_t)255;
        return p;
    };
    float* deg  = carve((size_t)N_NODES);            // degrees -> dinv (in place)
    float* h1   = carve((size_t)N_NODES * HID);      // x@W1+b1
    float* agg1 = carve((size_t)N_NODES * HID);      // aggregated, then relu'd in place
    float* h2   = carve((size_t)N_NODES * N_CLS);    // h1a@W2+b2
    float* agg2 = carve((size_t)N_NODES * N_CLS);    // aggregated layer 2

    const int B = 256;
    auto blocks = [](long long n, int b) { return (unsigned)((n + b - 1) / b); };

    // zero accumulation buffers (must re-init every call; harness doesn't)
    k_zero<<<blocks((long long)N_NODES * HID, B), B, 0, stream>>>(agg1, N_NODES * HID);
    k_zero<<<blocks((long long)N_NODES * N_CLS, B), B, 0, stream>>>(agg2, N_NODES * N_CLS);

    // degrees (self loop = 1) -> dinv
    k_init_deg<<<blocks(N_NODES, B), B, 0, stream>>>(deg, N_NODES);
    k_deg_accum<<<blocks(N_EDGES, B), B, 0, stream>>>(col, deg, N_EDGES);
    k_dinv<<<blocks(N_NODES, B), B, 0, stream>>>(deg, N_NODES);

    // layer 1
    const long long g1_threads = (long long)(N_NODES / 16) * 32;   // one wave per 16-row tile
    k_gemm1_wmma<<<blocks(g1_threads, B), B, 0, stream>>>(x, W1, b1, h1);
    k_aggregate<HID, 16><<<blocks((long long)N_EDGES * 16, B), B, 0, stream>>>(
        row, col, deg, h1, agg1, N_EDGES);
    k_finalize1<<<blocks((long long)N_NODES * HID, B), B, 0, stream>>>(agg1, h1, deg);

    // layer 2
    k_gemm2<<<blocks(N_NODES, B), B, 0, stream>>>(agg1, W2, b2, h2);
    k_aggregate<N_CLS, 8><<<blocks((long long)N_EDGES * 8, B), B, 0, stream>>>(
        row, col, deg, h2, agg2, N_EDGES);
    k_final<<<blocks(N_NODES, B), B, 0, stream>>>(agg2, h2, deg, out);
}